// ConnectivityLoss_41377714929670
// MI455X (gfx1250) — compile-verified
//
#include <hip/hip_runtime.h>
#include <math.h>

// Problem constants (from reference setup_inputs).
#define Bn   8
#define Hn   768
#define Wn   768
#define HWn  (Hn * Wn)          // 589824; /256 = 2304, /1024 = 576 (exact)
#define PEN  10.0f              // PENALTY; ALPHA == 1.0 so loss = weighted mean

typedef __attribute__((ext_vector_type(2))) float v2f;
typedef __attribute__((ext_vector_type(8))) float v8f;

// ---------------- low-level helpers ----------------

__device__ __forceinline__ int ld_agent(const int* p) {
    // Relaxed agent-scope load: must observe L2-resident atomicMin updates
    // from other WGPs while the union-find is being mutated.
    return __hip_atomic_load(p, __ATOMIC_RELAXED, __HIP_MEMORY_SCOPE_AGENT);
}

__device__ __forceinline__ int uf_find(int* L, int a) {
    int c = ld_agent(&L[a]);
    while (c != a) { a = c; c = ld_agent(&L[a]); }
    return a;
}

// Playne-Stephens style union via atomicMin with retry on the returned value.
__device__ __forceinline__ void uf_union(int* L, int a, int b) {
    while (true) {
        a = uf_find(L, a);
        b = uf_find(L, b);
        if (a == b) return;
        if (a < b) { int old = atomicMin(&L[b], a); if (old == b) return; b = old; }
        else       { int old = atomicMin(&L[a], b); if (old == a) return; a = old; }
    }
}

// One ds_swizzle xor-step within groups of 16 (and_mask=0x1f, or=0).
#define SWZ_ADD(s, xorm) \
    s += __int_as_float(__builtin_amdgcn_ds_swizzle(__float_as_int(s), ((xorm) << 10) | 0x1f))

// Wave32 sum of 2 partials/lane via V_WMMA_F32_16X16X4_F32:
// A = ones(16x4), B(4x16) carries the 64 partials. D[m][n] = sum_k B[k][n],
// so every lane's d[0] is a column sum; lanes 0..15 hold s[0..15] and lanes
// 16..31 duplicate them (row M=8). A 4-step xor-reduce within each group of
// 16 therefore leaves the exact wave total in every lane.
__device__ __forceinline__ float wave_sum_wmma(float p01, float p23) {
    v2f a; a[0] = 1.0f; a[1] = 1.0f;
    v2f b; b[0] = p01;  b[1] = p23;
    v8f c = {};
    c = __builtin_amdgcn_wmma_f32_16x16x4_f32(false, a, false, b,
                                              (short)0, c, false, false);
    float s = c[0];
    SWZ_ADD(s, 1);
    SWZ_ADD(s, 2);
    SWZ_ADD(s, 4);
    SWZ_ADD(s, 8);
    return s;
}

// Branch-free per-pixel weighted L1 term.
__device__ __forceinline__ float px_loss(float pr, float tg, bool neg, int pl,
                                         const int* __restrict__ smin,
                                         const int* __restrict__ smax) {
    float pp  = neg ? (pr + 1.0f) * 0.5f : pr;
    float t01 = (tg + 1.0f) * 0.5f;
    int   idx = (pl >= 0) ? pl : 0;         // keep gather loads unconditional
    int   lo  = smin[idx];
    int   hi  = smax[idx];
    bool  tm  = t01 > 0.5f;
    bool  err = (pl >= 0) & (!tm) & (hi > lo);
    float w   = err ? (1.0f + PEN) : 1.0f;
    return fabsf(pp - t01) * w;
}

// ---------------- kernels ----------------

__global__ void k_zero(int* flag, float* accum) {
    if (threadIdx.x == 0) { *flag = 0; *accum = 0.0f; }
}

// flag = any(pred < 0) over the whole batch (reference: jnp.min(pred) < 0).
__global__ void k_negscan(const float4* __restrict__ pred4, int* flag) {
    int i = blockIdx.x * 256 + threadIdx.x;
    float4 v = pred4[i];
    int n = (v.x < 0.0f) | (v.y < 0.0f) | (v.z < 0.0f) | (v.w < 0.0f);
    if (__syncthreads_or(n) && threadIdx.x == 0) atomicOr(flag, 1);
}

// Per-image: build masks as union-find seeds, init segment-min/max arrays.
__global__ void k_init(const float4* __restrict__ pred4,
                       const float4* __restrict__ targ4,
                       const int* __restrict__ flag,
                       int4* plab, int4* tlab, int4* smin, int4* smax) {
    int i = blockIdx.x * 256 + threadIdx.x;
    float4 pr = pred4[i], tg = targ4[i];
    bool neg = (*flag) != 0;
    int base = i * 4;
    int4 pl, tl;
    #define PMASK(v) ((neg ? ((v) + 1.0f) * 0.5f : (v)) > 0.5f)
    #define TMASK(v) ((((v) + 1.0f) * 0.5f) > 0.5f)
    pl.x = PMASK(pr.x) ? base + 0 : -1;
    pl.y = PMASK(pr.y) ? base + 1 : -1;
    pl.z = PMASK(pr.z) ? base + 2 : -1;
    pl.w = PMASK(pr.w) ? base + 3 : -1;
    tl.x = TMASK(tg.x) ? base + 0 : -1;
    tl.y = TMASK(tg.y) ? base + 1 : -1;
    tl.z = TMASK(tg.z) ? base + 2 : -1;
    tl.w = TMASK(tg.w) ? base + 3 : -1;
    #undef PMASK
    #undef TMASK
    plab[i] = pl;
    tlab[i] = tl;
    smin[i] = make_int4(0x7fffffff, 0x7fffffff, 0x7fffffff, 0x7fffffff);
    smax[i] = make_int4(-1, -1, -1, -1);
}

// 4-connectivity: union with left and up neighbors (covers all edges).
__global__ void k_merge(int* L) {
    int i = blockIdx.x * 256 + threadIdx.x;
    if (ld_agent(&L[i]) < 0) return;
    int x = i % Wn;
    if (x > 0   && ld_agent(&L[i - 1])  >= 0) uf_union(L, i, i - 1);
    if (i >= Wn && ld_agent(&L[i - Wn]) >= 0) uf_union(L, i, i - Wn);
}

// Replace each foreground label with its root (root index == component id).
__global__ void k_flatten(int* L) {
    int i = blockIdx.x * 256 + threadIdx.x;
    if (ld_agent(&L[i]) >= 0) L[i] = uf_find(L, i);
}

// smin/smax over target component-ids per pred component (indexed by root).
__global__ void k_segminmax(const int4* __restrict__ plab4,
                            const int4* __restrict__ tlab4,
                            int* smin, int* smax) {
    int i = blockIdx.x * 256 + threadIdx.x;
    int4 pl = plab4[i];
    int4 tl = tlab4[i];
    if (pl.x >= 0 && tl.x >= 0) { atomicMin(&smin[pl.x], tl.x); atomicMax(&smax[pl.x], tl.x); }
    if (pl.y >= 0 && tl.y >= 0) { atomicMin(&smin[pl.y], tl.y); atomicMax(&smax[pl.y], tl.y); }
    if (pl.z >= 0 && tl.z >= 0) { atomicMin(&smin[pl.z], tl.z); atomicMax(&smax[pl.z], tl.z); }
    if (pl.w >= 0 && tl.w >= 0) { atomicMin(&smin[pl.w], tl.w); atomicMax(&smax[pl.w], tl.w); }
}

// Weighted L1 with merge-penalty; 4 pixels/lane, WMMA wave reduction.
// Uniform control flow end-to-end => EXEC all-ones at the WMMA.
__global__ void k_loss(const float4* __restrict__ pred4,
                       const float4* __restrict__ targ4,
                       const int* __restrict__ flag,
                       const int4* __restrict__ plab4,
                       const int* __restrict__ smin,
                       const int* __restrict__ smax,
                       float* accum) {
    int i = blockIdx.x * 256 + threadIdx.x;
    float4 pr = pred4[i], tg = targ4[i];
    int4 pl = plab4[i];
    bool neg = (*flag) != 0;

    float l0 = px_loss(pr.x, tg.x, neg, pl.x, smin, smax);
    float l1 = px_loss(pr.y, tg.y, neg, pl.y, smin, smax);
    float l2 = px_loss(pr.z, tg.z, neg, pl.z, smin, smax);
    float l3 = px_loss(pr.w, tg.w, neg, pl.w, smin, smax);

    float wsum = wave_sum_wmma(l0 + l1, l2 + l3);   // 128 pixels per wave
    if ((threadIdx.x & 31) == 0) atomicAdd(accum, wsum);
}

__global__ void k_final(const float* __restrict__ accum, float* out) {
    if (threadIdx.x == 0)
        out[0] = accum[0] * (1.0f / (float)((long long)Bn * HWn));
}

// ---------------- launch ----------------

extern "C" void kernel_launch(void* const* d_in, const int* in_sizes, int n_in,
                              void* d_out, int out_size, void* d_ws, size_t ws_size,
                              hipStream_t stream) {
    const float* pred = (const float*)d_in[0];
    const float* targ = (const float*)d_in[1];
    float* out = (float*)d_out;

    char* ws = (char*)d_ws;                 // ~9.44 MB used, reused per image
    int*   plab  = (int*)(ws);
    int*   tlab  = (int*)(ws + (size_t)HWn * 4);
    int*   smin  = (int*)(ws + (size_t)HWn * 8);
    int*   smax  = (int*)(ws + (size_t)HWn * 12);
    int*   flag  = (int*)(ws + (size_t)HWn * 16);
    float* accum = (float*)(ws + (size_t)HWn * 16 + 8);

    dim3 blk(256);                          // 8 waves / block on wave32
    dim3 g1(HWn / 256);                     // scalar kernels: 2304 blocks
    dim3 g4(HWn / 1024);                    // float4 kernels:  576 blocks
    dim3 gB4((Bn * HWn) / 1024);            // batch float4:   4608 blocks

    k_zero<<<1, 32, 0, stream>>>(flag, accum);
    k_negscan<<<gB4, blk, 0, stream>>>((const float4*)pred, flag);

    for (int b = 0; b < Bn; ++b) {
        const float* pb = pred + (size_t)b * HWn;
        const float* tb = targ + (size_t)b * HWn;
        k_init<<<g4, blk, 0, stream>>>((const float4*)pb, (const float4*)tb, flag,
                                       (int4*)plab, (int4*)tlab,
                                       (int4*)smin, (int4*)smax);
        k_merge<<<g1, blk, 0, stream>>>(plab);
        k_merge<<<g1, blk, 0, stream>>>(tlab);
        k_flatten<<<g1, blk, 0, stream>>>(plab);
        k_flatten<<<g1, blk, 0, stream>>>(tlab);
        k_segminmax<<<g4, blk, 0, stream>>>((const int4*)plab, (const int4*)tlab,
                                            smin, smax);
        k_loss<<<g4, blk, 0, stream>>>((const float4*)pb, (const float4*)tb, flag,
                                       (const int4*)plab, smin, smax, accum);
    }

    k_final<<<1, 32, 0, stream>>>(accum, out);
}